// AssetGATEncoder_28217935135373
// MI455X (gfx1250) — compile-verified
//
#include <hip/hip_runtime.h>
#include <hip/hip_bf16.h>

typedef __attribute__((ext_vector_type(2))) float v2f;
typedef __attribute__((ext_vector_type(8))) float v8f;

#define SLOPE 0.2f
#define SM_EPS 1e-16f
#define LN_EPS 1e-5f

__device__ __forceinline__ unsigned encodeOrdered(float f) {
    unsigned u = __float_as_uint(f);
    return (u & 0x80000000u) ? ~u : (u | 0x80000000u);
}
__device__ __forceinline__ float decodeOrdered(unsigned e) {
    return __uint_as_float((e & 0x80000000u) ? (e & 0x7fffffffu) : ~e);
}
__device__ __forceinline__ float eluf(float v) {
    return v > 0.f ? v : (__expf(v) - 1.f);
}
__device__ __forceinline__ float lrelu(float v) {
    return v > 0.f ? v : SLOPE * v;
}

// ---------------------------------------------------------------------------
// Generic fp32 WMMA GEMM: C[M,N] = act(A[M,K] @ B[K,N] + bias[N])
// One wave per 16x16 output tile; 4 waves per block over consecutive M tiles.
// A-frag (16x4 f32, 2 VGPRs): lanes 0-15 -> K = k0+{0,1}; lanes 16-31 -> K = k0+{2,3}
// B-frag (4x16 f32, 2 VGPRs): mirrored layout (N striped across lane halves)
// C/D (16x16 f32, 8 VGPRs): VGPR r, lanes 0-15 -> M=r; lanes 16-31 -> M=r+8
// ---------------------------------------------------------------------------
__global__ __launch_bounds__(128) void gemm16_wmma(
    const float* __restrict__ A, const float* __restrict__ B,
    const float* __restrict__ bias, float* __restrict__ C,
    int Mtiles, int N, int K, int act)
{
    int mt = blockIdx.x * 4 + (threadIdx.x >> 5);
    if (mt >= Mtiles) return;
    int nt = blockIdx.y;
    int lane = threadIdx.x & 31;
    int half = lane >> 4;
    int l = lane & 15;

    const float* Arow = A + (size_t)(mt * 16 + l) * K;
    const float* Bcol = B + (size_t)(nt * 16 + l);

    v8f acc = {};
    for (int k0 = 0; k0 < K; k0 += 4) {
        int ka = k0 + 2 * half;
        v2f a, b;
        a.x = Arow[ka];
        a.y = Arow[ka + 1];
        b.x = Bcol[(size_t)ka * N];
        b.y = Bcol[(size_t)(ka + 1) * N];
        acc = __builtin_amdgcn_wmma_f32_16x16x4_f32(
            false, a, false, b, (short)0, acc, false, false);
    }

    float bv = bias[nt * 16 + l];
    float* Crow = C + (size_t)(mt * 16 + 8 * half) * N + nt * 16 + l;
#pragma unroll
    for (int r = 0; r < 8; ++r) {
        float v = acc[r] + bv;
        if (act == 1) v = eluf(v);
        Crow[(size_t)r * N] = v;
    }
}

// ---------------------------------------------------------------------------
__global__ void zero_f(float* __restrict__ p, long n)
{
    long t = (long)blockIdx.x * blockDim.x + threadIdx.x;
    if (t < n) p[t] = 0.f;
}

// ---------------------------------------------------------------------------
// Layer 0 edge score (4 heads x 64 dims). Wave per edge, lane strides dims.
// ---------------------------------------------------------------------------
__global__ __launch_bounds__(128) void edge_score0(
    const float* __restrict__ XL, const float* __restrict__ XR,
    const float* __restrict__ att, const int* __restrict__ src,
    const int* __restrict__ dst, float* __restrict__ escore,
    unsigned* __restrict__ mEnc, int E, int Etot)
{
    int e = blockIdx.x * 4 + (threadIdx.x >> 5);
    if (e >= Etot) return;
    int lane = threadIdx.x & 31;
    int s, d;
    if (e < E) { s = src[e]; d = dst[e]; } else { s = e - E; d = s; }
    const float* xls = XL + (size_t)s * 256;
    const float* xrd = XR + (size_t)d * 256;

    float acc0, acc1, acc2, acc3;
    float accs[4];
#pragma unroll
    for (int h = 0; h < 4; ++h) {
        int i0 = h * 64 + lane, i1 = i0 + 32;
        float z0 = lrelu(xls[i0] + xrd[i0]);
        float z1 = lrelu(xls[i1] + xrd[i1]);
        accs[h] = z0 * att[i0] + z1 * att[i1];
    }
    acc0 = accs[0]; acc1 = accs[1]; acc2 = accs[2]; acc3 = accs[3];
#pragma unroll
    for (int off = 16; off >= 1; off >>= 1) {
        acc0 += __shfl_xor(acc0, off, 32);
        acc1 += __shfl_xor(acc1, off, 32);
        acc2 += __shfl_xor(acc2, off, 32);
        acc3 += __shfl_xor(acc3, off, 32);
    }
    if (lane < 4) {
        float v = (lane == 0) ? acc0 : (lane == 1) ? acc1 : (lane == 2) ? acc2 : acc3;
        escore[(size_t)e * 4 + lane] = v;
        atomicMax(&mEnc[(size_t)d * 4 + lane], encodeOrdered(v));
    }
}

// a = exp(e - m[dst]); in-place over escore; atomic sum into S
__global__ void exp_sum0(
    float* __restrict__ escore, const unsigned* __restrict__ mEnc,
    float* __restrict__ S, const int* __restrict__ dst, int E, int Etot)
{
    long t = (long)blockIdx.x * blockDim.x + threadIdx.x;
    if (t >= (long)Etot * 4) return;
    int e = (int)(t >> 2), h = (int)(t & 3);
    int d = (e < E) ? dst[e] : e - E;
    float a = __expf(escore[t] - decodeOrdered(mEnc[d * 4 + h]));
    escore[t] = a;
    atomicAdd(&S[d * 4 + h], a);
}

__global__ __launch_bounds__(128) void scatter0(
    const float* __restrict__ XL, const float* __restrict__ a,
    const float* __restrict__ S, const int* __restrict__ src,
    const int* __restrict__ dst, float* __restrict__ out, int E, int Etot)
{
    int e = blockIdx.x * 4 + (threadIdx.x >> 5);
    if (e >= Etot) return;
    int lane = threadIdx.x & 31;
    int s, d;
    if (e < E) { s = src[e]; d = dst[e]; } else { s = e - E; d = s; }
    const float* xls = XL + (size_t)s * 256;
    float* od = out + (size_t)d * 256;
#pragma unroll
    for (int h = 0; h < 4; ++h) {
        float alpha = a[(size_t)e * 4 + h] / (S[d * 4 + h] + SM_EPS);
        int i0 = h * 64 + lane;
        atomicAdd(&od[i0], alpha * xls[i0]);
        atomicAdd(&od[i0 + 32], alpha * xls[i0 + 32]);
    }
}

// LN(x + bias) * g + be, then ELU; in place; wave per row of 256
__global__ __launch_bounds__(128) void ln_elu_256(
    float* __restrict__ H, const float* __restrict__ bias,
    const float* __restrict__ g, const float* __restrict__ be, int Nrows)
{
    int row = blockIdx.x * 4 + (threadIdx.x >> 5);
    if (row >= Nrows) return;
    int lane = threadIdx.x & 31;
    float* hr = H + (size_t)row * 256;
    float v[8];
    float sum = 0.f;
#pragma unroll
    for (int i = 0; i < 8; ++i) { v[i] = hr[lane + 32 * i] + bias[lane + 32 * i]; sum += v[i]; }
#pragma unroll
    for (int off = 16; off >= 1; off >>= 1) sum += __shfl_xor(sum, off, 32);
    float mu = sum * (1.f / 256.f);
    float var = 0.f;
#pragma unroll
    for (int i = 0; i < 8; ++i) { float t = v[i] - mu; var += t * t; }
#pragma unroll
    for (int off = 16; off >= 1; off >>= 1) var += __shfl_xor(var, off, 32);
    float rs = rsqrtf(var * (1.f / 256.f) + LN_EPS);
#pragma unroll
    for (int i = 0; i < 8; ++i) {
        float t = (v[i] - mu) * rs * g[lane + 32 * i] + be[lane + 32 * i];
        hr[lane + 32 * i] = eluf(t);
    }
}

// ---------------------------------------------------------------------------
// Layer 1 (1 head, 32 dims). Wave per edge, one element per lane.
// ---------------------------------------------------------------------------
__global__ __launch_bounds__(128) void edge_score1(
    const float* __restrict__ XL, const float* __restrict__ XR,
    const float* __restrict__ att, const int* __restrict__ src,
    const int* __restrict__ dst, float* __restrict__ escore,
    unsigned* __restrict__ mEnc, int E, int Etot)
{
    int e = blockIdx.x * 4 + (threadIdx.x >> 5);
    if (e >= Etot) return;
    int lane = threadIdx.x & 31;
    int s, d;
    if (e < E) { s = src[e]; d = dst[e]; } else { s = e - E; d = s; }
    float z = lrelu(XL[(size_t)s * 32 + lane] + XR[(size_t)d * 32 + lane]);
    float acc = z * att[lane];
#pragma unroll
    for (int off = 16; off >= 1; off >>= 1) acc += __shfl_xor(acc, off, 32);
    if (lane == 0) {
        escore[e] = acc;
        atomicMax(&mEnc[d], encodeOrdered(acc));
    }
}

__global__ void exp_sum1(
    float* __restrict__ escore, const unsigned* __restrict__ mEnc,
    float* __restrict__ S, const int* __restrict__ dst, int E, int Etot)
{
    int e = blockIdx.x * blockDim.x + threadIdx.x;
    if (e >= Etot) return;
    int d = (e < E) ? dst[e] : e - E;
    float a = __expf(escore[e] - decodeOrdered(mEnc[d]));
    escore[e] = a;
    atomicAdd(&S[d], a);
}

__global__ __launch_bounds__(128) void scatter1(
    const float* __restrict__ XL, const float* __restrict__ a,
    const float* __restrict__ S, const int* __restrict__ src,
    const int* __restrict__ dst, float* __restrict__ out, int E, int Etot)
{
    int e = blockIdx.x * 4 + (threadIdx.x >> 5);
    if (e >= Etot) return;
    int lane = threadIdx.x & 31;
    int s, d;
    if (e < E) { s = src[e]; d = dst[e]; } else { s = e - E; d = s; }
    float alpha = a[e] / (S[d] + SM_EPS);
    atomicAdd(&out[(size_t)d * 32 + lane], alpha * XL[(size_t)s * 32 + lane]);
}

// Final LN over 32 dims, write to d_out
__global__ __launch_bounds__(128) void ln_32_out(
    const float* __restrict__ H, const float* __restrict__ bias,
    const float* __restrict__ g, const float* __restrict__ be,
    float* __restrict__ out, int Nrows)
{
    int row = blockIdx.x * 4 + (threadIdx.x >> 5);
    if (row >= Nrows) return;
    int lane = threadIdx.x & 31;
    float v = H[(size_t)row * 32 + lane] + bias[lane];
    float sum = v;
#pragma unroll
    for (int off = 16; off >= 1; off >>= 1) sum += __shfl_xor(sum, off, 32);
    float mu = sum * (1.f / 32.f);
    float t = v - mu;
    float var = t * t;
#pragma unroll
    for (int off = 16; off >= 1; off >>= 1) var += __shfl_xor(var, off, 32);
    float rs = rsqrtf(var * (1.f / 32.f) + LN_EPS);
    out[(size_t)row * 32 + lane] = t * rs * g[lane] + be[lane];
}

// ---------------------------------------------------------------------------
extern "C" void kernel_launch(void* const* d_in, const int* in_sizes, int n_in,
                              void* d_out, int out_size, void* d_ws, size_t ws_size,
                              hipStream_t stream)
{
    const float* x    = (const float*)d_in[0];
    const int*   ei   = (const int*)  d_in[1];
    const float* Wp   = (const float*)d_in[2];
    const float* bp   = (const float*)d_in[3];
    const float* Wl0  = (const float*)d_in[4];
    const float* bl0  = (const float*)d_in[5];
    const float* Wr0  = (const float*)d_in[6];
    const float* br0  = (const float*)d_in[7];
    const float* att0 = (const float*)d_in[8];
    const float* bias0= (const float*)d_in[9];
    const float* g0   = (const float*)d_in[10];
    const float* be0  = (const float*)d_in[11];
    const float* Wl1  = (const float*)d_in[12];
    const float* bl1  = (const float*)d_in[13];
    const float* Wr1  = (const float*)d_in[14];
    const float* br1  = (const float*)d_in[15];
    const float* att1 = (const float*)d_in[16];
    const float* bias1= (const float*)d_in[17];
    const float* g1   = (const float*)d_in[18];
    const float* be1  = (const float*)d_in[19];

    const int Nn   = in_sizes[0] / 128;   // 30000 nodes
    const int E    = in_sizes[1] / 2;     // 480000 edges
    const int Etot = E + Nn;              // + self loops
    const int* src = ei;
    const int* dst = ei + E;
    const int Mt = Nn / 16;               // 1875 full M tiles

    // workspace layout (floats)
    float* ws  = (float*)d_ws;
    float* H0  = ws;                                 // [Nn,64]  -> reused: XL1/XR1
    float* XL0 = H0  + (size_t)Nn * 64;              // [Nn,256] -> reused: out1
    float* XR0 = XL0 + (size_t)Nn * 256;             // [Nn,256] -> reused: out0 / h2
    float* ES  = XR0 + (size_t)Nn * 256;             // [Etot,4] score / a (both layers)
    unsigned* M0 = (unsigned*)(ES + (size_t)Etot * 4);  // [Nn,4]
    float* S0  = (float*)(M0 + (size_t)Nn * 4);      // [Nn,4]  (contiguous after M0)
    float* XL1 = H0;                                 // [Nn,32]
    float* XR1 = H0 + (size_t)Nn * 32;               // [Nn,32]
    float* OUT1 = XL0;                               // [Nn,32]

    const int eb = (Etot + 3) / 4;       // wave-per-edge blocks (4 waves/block)
    const int gm = (Mt + 3) / 4;         // gemm blocks along M

    // 1) H0 = ELU(x @ Wp + bp)          [Nn,128]x[128,64]
    gemm16_wmma<<<dim3(gm, 64 / 16), 128, 0, stream>>>(x, Wp, bp, H0, Mt, 64, 128, 1);
    // 2) XL0 / XR0                      [Nn,64]x[64,256]
    gemm16_wmma<<<dim3(gm, 256 / 16), 128, 0, stream>>>(H0, Wl0, bl0, XL0, Mt, 256, 64, 0);
    gemm16_wmma<<<dim3(gm, 256 / 16), 128, 0, stream>>>(H0, Wr0, br0, XR0, Mt, 256, 64, 0);
    // 3) init max(encoded)/sum (contiguous region M0|S0 = 2*Nn*4 words)
    zero_f<<<((long)Nn * 8 + 255) / 256, 256, 0, stream>>>((float*)M0, (long)Nn * 8);
    // 4) per-edge scores + segment max
    edge_score0<<<eb, 128, 0, stream>>>(XL0, XR0, att0, src, dst, ES, M0, E, Etot);
    // 5) out0 accumulator (overwrites XR0 -- XR0 dead after scoring)
    zero_f<<<((long)Nn * 256 + 255) / 256, 256, 0, stream>>>(XR0, (long)Nn * 256);
    // 6) exp + segment sum
    exp_sum0<<<((long)Etot * 4 + 255) / 256, 256, 0, stream>>>(ES, M0, S0, dst, E, Etot);
    // 7) alpha-weighted scatter
    scatter0<<<eb, 128, 0, stream>>>(XL0, ES, S0, src, dst, XR0, E, Etot);
    // 8) h2 = ELU(LN(out0 + bias0))  (in place in XR0)
    ln_elu_256<<<(Nn + 3) / 4, 128, 0, stream>>>(XR0, bias0, g0, be0, Nn);
    // 9) XL1 / XR1                      [Nn,256]x[256,32]
    gemm16_wmma<<<dim3(gm, 32 / 16), 128, 0, stream>>>(XR0, Wl1, bl1, XL1, Mt, 32, 256, 0);
    gemm16_wmma<<<dim3(gm, 32 / 16), 128, 0, stream>>>(XR0, Wr1, br1, XR1, Mt, 32, 256, 0);
    // 10) re-init max/sum (only first Nn entries used), zero out1
    zero_f<<<((long)Nn * 8 + 255) / 256, 256, 0, stream>>>((float*)M0, (long)Nn * 8);
    zero_f<<<((long)Nn * 32 + 255) / 256, 256, 0, stream>>>(OUT1, (long)Nn * 32);
    // 11) layer-1 edge phase
    edge_score1<<<eb, 128, 0, stream>>>(XL1, XR1, att1, src, dst, ES, M0, E, Etot);
    exp_sum1<<<(Etot + 255) / 256, 256, 0, stream>>>(ES, M0, S0, dst, E, Etot);
    scatter1<<<eb, 128, 0, stream>>>(XL1, ES, S0, src, dst, OUT1, E, Etot);
    // 12) final LN -> d_out
    ln_32_out<<<(Nn + 3) / 4, 128, 0, stream>>>(OUT1, bias1, g1, be1, (float*)d_out, Nn);
}